// Loss_71837622992952
// MI455X (gfx1250) — compile-verified
//
#include <hip/hip_runtime.h>
#include <hip/hip_bf16.h>
#include <math.h>

typedef __attribute__((ext_vector_type(2))) float v2f;
typedef __attribute__((ext_vector_type(8))) float v8f;

#define WAVE 32
#define NCLS 13

// ---------------- helpers ----------------

__device__ __forceinline__ unsigned int sortable_f32(float f) {
  unsigned int b = __float_as_uint(f);
  return b ^ (((int)b < 0) ? 0xFFFFFFFFu : 0x80000000u);
}

__device__ __forceinline__ unsigned long long min_u64(unsigned long long a, unsigned long long b) {
  return a < b ? a : b;
}

// 64-bit shfl_xor via two 32-bit shuffles (wave32-safe)
__device__ __forceinline__ unsigned long long shflxor_u64(unsigned long long v, int mask, int width) {
  unsigned int lo = (unsigned int)v;
  unsigned int hi = (unsigned int)(v >> 32);
  lo = __shfl_xor(lo, mask, width);
  hi = __shfl_xor(hi, mask, width);
  return ((unsigned long long)hi << 32) | lo;
}

__device__ __forceinline__ float wave_sum(float s) {
#pragma unroll
  for (int off = 16; off; off >>= 1) s += __shfl_xor(s, off, WAVE);
  return s;
}

// WMMA A operand: 16 queries (K=3 padded to 4).
// A layout (16x4 f32): lanes 0-15 -> (K0,K1)=(x,y); lanes 16-31 -> (K2,K3)=(z,0).
__device__ __forceinline__ v2f make_A(const float* qxyz) {
  int lane = threadIdx.x;
  int half = lane >> 4, lm = lane & 15;
  const float* q = qxyz + lm * 3;
  v2f a;
  a.x = half ? q[2] : q[0];
  a.y = half ? 0.0f : q[1];
  return a;
}

// ---------------- init / finalize ----------------

__global__ void init_acc_kernel(float* acc) {
  if (threadIdx.x < 16) acc[threadIdx.x] = 0.0f;
}

__global__ void finalize_kernel(const float* acc, float* out) {
  const int Nsl[5] = {8192, 2048, 512, 128, 32};
  float loss = acc[0] / (2.0f * 8192.0f);              // CE(output) mean
  float bl = 0.0f;
  for (int i = 0; i < 5; ++i) bl += acc[6 + i] / fmaxf(acc[11 + i], 1.0f);
  float ol = 0.0f;
  for (int j = 0; j < 5; ++j) ol += acc[1 + j] / (2.0f * (float)Nsl[j]);
  out[0] = loss + 0.1f * bl + 0.1f * ol;               // + ALPHA*b + BETA*o
}

// ---------------- cross entropy ----------------
// rows = B*Np, logits [rows, 13]; label = gidx ? labels[b*Nfull + gidx[row]] : labels[row]
__global__ void ce_kernel(const float* __restrict__ logits, const int* __restrict__ labels,
                          const int* __restrict__ gidx, int Np, int Nfull, int rows,
                          float* __restrict__ acc) {
  int tid = blockIdx.x * blockDim.x + threadIdx.x;
  float contrib = 0.0f;
  if (tid < rows) {
    int b = tid / Np;
    int lab = gidx ? labels[(size_t)b * Nfull + gidx[tid]] : labels[tid];
    const float* lr = logits + (size_t)tid * NCLS;
    float mx = lr[0];
#pragma unroll
    for (int c = 1; c < NCLS; ++c) mx = fmaxf(mx, lr[c]);
    float se = 0.0f;
#pragma unroll
    for (int c = 0; c < NCLS; ++c) se += expf(lr[c] - mx);
    contrib = logf(se) + mx - lr[lab];
  }
  contrib = wave_sum(contrib);                 // one atomic per wave
  if ((threadIdx.x & (WAVE - 1)) == 0) atomicAdd(acc, contrib);
}

// ---------------- KNN soft labels ----------------
// One wave per 16-query tile. kn rounds of threshold-scan selection over Nc candidates,
// dot products via V_WMMA_F32_16X16X4_F32. The cross-lane min runs ONCE per round
// (per-lane mins accumulated over the whole candidate scan), keeping LDS traffic tiny.
__global__ void __launch_bounds__(32) knn_soft_kernel(
    const float* __restrict__ q_xyz,   // [B, Nq, 3]
    const float* __restrict__ c_xyz,   // [B, Nc, 3]
    const int* __restrict__ labels,    // [B, Nc]
    int Nq, int Nc, int kn,
    float* __restrict__ soft)          // [B, Nq, 13]
{
  __shared__ int cnt[16][NCLS];
  int lane = threadIdx.x;
  int half = lane >> 4, lm = lane & 15;
  int b = blockIdx.y, tile = blockIdx.x;

  for (int i = lane; i < 16 * NCLS; i += WAVE) (&cnt[0][0])[i] = 0;
  __syncthreads();

  v2f A = make_A(q_xyz + ((size_t)b * Nq + tile * 16) * 3);
  const float* cb = c_xyz + (size_t)b * Nc * 3;
  const int* lb = labels + (size_t)b * Nc;

  unsigned long long lastk[8];
#pragma unroll
  for (int r = 0; r < 8; ++r) lastk[r] = 0ULL;

  for (int round = 0; round < kn; ++round) {
    unsigned long long best[8];
#pragma unroll
    for (int r = 0; r < 8; ++r) best[r] = ~0ULL;

    for (int t = 0; t < Nc; t += 16) {
      int n = t + lm;
      float cx = cb[n * 3 + 0], cy = cb[n * 3 + 1], cz = cb[n * 3 + 2];
      v2f Bv;
      Bv.x = half ? cz : cx;
      Bv.y = half ? 0.0f : cy;
      float xn = cx * cx + cy * cy + cz * cz;   // |x|^2 (|q|^2 constant per query: drop)
      v8f C = {};
      C = __builtin_amdgcn_wmma_f32_16x16x4_f32(false, A, false, Bv, (short)0, C, false, false);
#pragma unroll
      for (int r = 0; r < 8; ++r) {
        float v = fmaf(-2.0f, C[r], xn);
        unsigned long long key =
            ((unsigned long long)sortable_f32(v) << 32) | (unsigned int)n;
        key = (key > lastk[r]) ? key : ~0ULL;   // only keys strictly beyond last pick
        best[r] = min_u64(best[r], key);        // per-lane accumulation, no shuffles here
      }
    }
    // one cross-lane reduction per round (16-lane groups hold one M-row set each)
#pragma unroll
    for (int r = 0; r < 8; ++r) {
      unsigned long long k = best[r];
#pragma unroll
      for (int off = 8; off; off >>= 1) k = min_u64(k, shflxor_u64(k, off, 16));
      lastk[r] = k;
      if (lm == 0 && k != ~0ULL) {
        int sel = (int)(k & 0xFFFFFFFFu);
        cnt[half * 8 + r][lb[sel]] += 1;        // two lanes, distinct rows: no conflict
      }
    }
  }
  __syncthreads();

  float inv = 1.0f / (float)kn;
  for (int i = lane; i < 16 * NCLS; i += WAVE) {
    int m = i / NCLS, c = i % NCLS;
    soft[(((size_t)b * Nq) + tile * 16 + m) * NCLS + c] = (float)cnt[m][c] * inv;
  }
}

// ---------------- boundary loss ----------------
// Same WMMA threshold-scan KNN (queries==candidates==stage coords, ns rounds, drop self),
// then feature-space softmax contrast: one neighbor per lane, float4 feature loads.
__global__ void __launch_bounds__(32) boundary_kernel(
    const float* __restrict__ xyz,   // [B, Ni, 3]
    const float* __restrict__ feat,  // [B, Ni, D]
    const float* __restrict__ soft,  // [B, Ni, 13]
    int Ni, int D, int ns,
    float* __restrict__ bsum, float* __restrict__ bcnt)
{
  __shared__ int nbr[16][36];
  __shared__ float sfq[512];
  __shared__ float ldist[36];
  __shared__ int lmask[36];

  int lane = threadIdx.x;
  int half = lane >> 4, lm = lane & 15;
  int b = blockIdx.y, tile = blockIdx.x;

  v2f A = make_A(xyz + ((size_t)b * Ni + tile * 16) * 3);
  const float* cb = xyz + (size_t)b * Ni * 3;

  unsigned long long lastk[8];
#pragma unroll
  for (int r = 0; r < 8; ++r) lastk[r] = 0ULL;

  for (int round = 0; round < ns; ++round) {
    unsigned long long best[8];
#pragma unroll
    for (int r = 0; r < 8; ++r) best[r] = ~0ULL;

    for (int t = 0; t < Ni; t += 16) {
      int n = t + lm;
      float cx = cb[n * 3 + 0], cy = cb[n * 3 + 1], cz = cb[n * 3 + 2];
      v2f Bv;
      Bv.x = half ? cz : cx;
      Bv.y = half ? 0.0f : cy;
      float xn = cx * cx + cy * cy + cz * cz;
      v8f C = {};
      C = __builtin_amdgcn_wmma_f32_16x16x4_f32(false, A, false, Bv, (short)0, C, false, false);
#pragma unroll
      for (int r = 0; r < 8; ++r) {
        float v = fmaf(-2.0f, C[r], xn);
        unsigned long long key =
            ((unsigned long long)sortable_f32(v) << 32) | (unsigned int)n;
        key = (key > lastk[r]) ? key : ~0ULL;
        best[r] = min_u64(best[r], key);
      }
    }
#pragma unroll
    for (int r = 0; r < 8; ++r) {
      unsigned long long k = best[r];
#pragma unroll
      for (int off = 8; off; off >>= 1) k = min_u64(k, shflxor_u64(k, off, 16));
      lastk[r] = k;
      if (lm == 0) nbr[half * 8 + r][round] = (int)(k & 0xFFFFFFFFu);
    }
  }
  __syncthreads();

  int nm1 = ns - 1;
  for (int m = 0; m < 16; ++m) {
    size_t gq = (size_t)b * Ni + tile * 16 + m;
    const float* sq = soft + gq * NCLS;
    int hard = 0;
    float bv = sq[0];
#pragma unroll
    for (int c = 1; c < NCLS; ++c) {
      float s = sq[c];
      if (s > bv) { bv = s; hard = c; }
    }
    // stage query feature row in LDS (read as float4 below)
    const float* fq = feat + gq * (size_t)D;
    for (int d = lane; d < D; d += WAVE) sfq[d] = fq[d];
    __syncthreads();

    // one neighbor per lane (ns-1 <= 35 -> at most 2 passes), shuffle-free SSD
    for (int jbase = 1; jbase < ns; jbase += WAVE) {
      int j = jbase + lane;
      if (j < ns) {
        int nn = nbr[m][j];
        const float* sn = soft + ((size_t)b * Ni + nn) * NCLS;
        int nh = 0;
        float nbv = sn[0];
#pragma unroll
        for (int c = 1; c < NCLS; ++c) {
          float s = sn[c];
          if (s > nbv) { nbv = s; nh = c; }
        }
        const float4* fn4 = (const float4*)(feat + ((size_t)b * Ni + nn) * (size_t)D);
        const float4* fq4 = (const float4*)sfq;
        float s = 0.0f;
        for (int d4 = 0; d4 < (D >> 2); ++d4) {
          float4 a4 = fq4[d4];
          float4 b4 = fn4[d4];
          float d0 = a4.x - b4.x, d1 = a4.y - b4.y, d2 = a4.z - b4.z, d3 = a4.w - b4.w;
          s += d0 * d0 + d1 * d1 + d2 * d2 + d3 * d3;
        }
        ldist[j] = sqrtf(s + 1e-6f);
        lmask[j] = (hard == nh);
      }
    }
    __syncthreads();
    if (lane == 0) {
      float mx = -3.4e38f;
      for (int j = 1; j < ns; ++j) mx = fmaxf(mx, -ldist[j]);
      float pos = 0.0f, neg = 0.0f;
      int c2 = 0;
      for (int j = 1; j < ns; ++j) {
        float e = expf(-ldist[j] - mx);       // TEMPERATURE = 1
        neg += e;
        if (lmask[j]) { pos += e; ++c2; }
      }
      if (c2 > 0 && c2 < nm1) {
        atomicAdd(bsum, -logf(pos / neg + 1e-6f));
        atomicAdd(bcnt, 1.0f);
      }
    }
    __syncthreads();
  }
}

// ---------------- launch ----------------

extern "C" void kernel_launch(void* const* d_in, const int* in_sizes, int n_in,
                              void* d_out, int out_size, void* d_ws, size_t ws_size,
                              hipStream_t stream) {
  (void)in_sizes; (void)n_in; (void)out_size; (void)ws_size;
  const int* labels = (const int*)d_in[0];
  const int* indexs[4] = {(const int*)d_in[1], (const int*)d_in[2],
                          (const int*)d_in[3], (const int*)d_in[4]};
  const float* output = (const float*)d_in[5];
  const float* preds[5]; const float* xyzs[5]; const float* feats[5];
  for (int j = 0; j < 5; ++j) {
    preds[j] = (const float*)d_in[6 + 3 * j];
    xyzs[j]  = (const float*)d_in[7 + 3 * j];
    feats[j] = (const float*)d_in[8 + 3 * j];
  }

  const int B = 2, N = 8192;
  const int Nsl[5]   = {8192, 2048, 512, 128, 32};
  const int Dd[5]    = {64, 128, 256, 512, 512};
  const int NSAMP[5] = {36, 24, 24, 24, 24};
  const int KN[5]    = {1, 4, 16, 64, 256};

  float* acc  = (float*)d_ws;                         // 16 accumulators
  float* soft = (float*)((char*)d_ws + 256);          // reused per stage (stream-ordered)

  init_acc_kernel<<<1, 32, 0, stream>>>(acc);

  ce_kernel<<<(B * N + 127) / 128, 128, 0, stream>>>(output, labels, nullptr, N, N, B * N,
                                                     acc + 0);
  for (int j = 0; j < 5; ++j) {
    ce_kernel<<<(B * Nsl[j] + 127) / 128, 128, 0, stream>>>(
        preds[j], labels, j == 0 ? nullptr : indexs[j - 1], Nsl[j], N, B * Nsl[j], acc + 1 + j);
  }

  for (int i = 0; i < 5; ++i) {
    knn_soft_kernel<<<dim3(Nsl[i] / 16, B), 32, 0, stream>>>(
        xyzs[i], xyzs[0], labels, Nsl[i], N, KN[i], soft);
    boundary_kernel<<<dim3(Nsl[i] / 16, B), 32, 0, stream>>>(
        xyzs[i], feats[i], soft, Nsl[i], Dd[i], NSAMP[i], acc + 6 + i, acc + 11 + i);
  }

  finalize_kernel<<<1, 1, 0, stream>>>(acc, (float*)d_out);
}